// HierarchicalAutoEncoderLayer_60790967108240
// MI455X (gfx1250) — compile-verified
//
#include <hip/hip_runtime.h>

// ---------------------------------------------------------------------------
// MoE-SAE forward for MI455X (gfx1250, wave32).
// Memory-bound problem (~0.54 GFLOP over ~67 MB of expert weights -> AI ~= 8),
// so the design goal is: read each active expert's W_enc/W_dec once from HBM,
// batch 16 tokens per expert, and run the math on V_WMMA_F32_16X16X4_F32.
// ---------------------------------------------------------------------------

typedef __attribute__((ext_vector_type(2))) float v2f;
typedef __attribute__((ext_vector_type(8))) float v8f;

#define NSAE 128   // experts
#define DD   128   // d_data
#define DE   512   // d_dict
#define BTOT 1024  // batch
#define CAP  256   // per-expert bucket capacity (mean occupancy is 16)

#define XPAD 132   // LDS row stride for x tile   (132 % 64 == 4 -> no bank conflicts)
#define APAD 516   // LDS row stride for acts tile(516 % 64 == 4 -> no bank conflicts)

// --- Kernel 1: zero the accumulator output and the routing counters ---------
__global__ void sae_init_kernel(float* __restrict__ out, int* __restrict__ cnt) {
    int i = blockIdx.x * blockDim.x + threadIdx.x;
    if (i < BTOT * DD) out[i] = 0.0f;
    if (i < NSAE) cnt[i] = 0;
}

// --- Kernel 2: bucket (token, gate) pairs by expert -------------------------
// gate has exactly K nonzeros per row (top-k routing); sum-combine at the end
// makes intra-bucket order irrelevant.
__global__ void sae_route_kernel(const float* __restrict__ gate,
                                 int* __restrict__ cnt,
                                 int* __restrict__ btok,
                                 float* __restrict__ bg) {
    int tok = blockIdx.x * blockDim.x + threadIdx.x;
    if (tok >= BTOT) return;
    const float* g = gate + (size_t)tok * NSAE;
    for (int e = 0; e < NSAE; ++e) {
        float gv = g[e];
        if (gv != 0.0f) {
            int slot = atomicAdd(&cnt[e], 1);
            if (slot < CAP) {
                btok[e * CAP + slot] = tok;
                bg[e * CAP + slot]   = gv;
            }
        }
    }
}

// --- Kernel 3: per-expert tiled encode+decode on fp32 WMMA ------------------
// blockIdx.x = expert, blockIdx.y = tile slice. 256 threads = 8 wave32.
//
// WMMA 16x16x4 f32 fragment layouts (ISA 7.12.2):
//   A (16x4, M on lanes):   vgpr v, lane L: A[M = L%16][K = v + 2*(L/16)]
//   B (4x16, N on lanes):   vgpr v, lane L: B[K = v + 2*(L/16)][N = L%16]
//   C/D (16x16):            vgpr r, lane L: C[M = r + 8*(L/16)][N = L%16]
__global__ __launch_bounds__(256, 1) void sae_tile_kernel(
    const float* __restrict__ x,      // [B, DD]
    const float* __restrict__ W_enc,  // [NSAE, DD, DE]
    const float* __restrict__ W_dec,  // [NSAE, DE, DD]
    const float* __restrict__ b_enc,  // [NSAE, DE]
    const float* __restrict__ b_dec,  // [NSAE, DD]
    const int*   __restrict__ cnt,
    const int*   __restrict__ btok,
    const float* __restrict__ bg,
    float* __restrict__ out)          // [B, DD], accumulated atomically
{
    __shared__ float xt[16 * XPAD];    // 16 tokens x 128 features
    __shared__ float acts[16 * APAD];  // 16 tokens x 512 activations
    __shared__ float gt[16];
    __shared__ int   tk[16];

    const int e = blockIdx.x;
    int n = cnt[e];
    if (n > CAP) n = CAP;
    const int ntiles = (n + 15) >> 4;

    const int tid  = threadIdx.x;
    const int wave = tid >> 5;        // 0..7
    const int lane = tid & 31;
    const int ln   = lane & 15;       // N / M index within fragment
    const int lh   = lane >> 4;       // half-wave select

    const float* We = W_enc + (size_t)e * DD * DE;
    const float* Wd = W_dec + (size_t)e * DE * DD;

    for (int t = blockIdx.y; t < ntiles; t += gridDim.y) {
        // ---- tile metadata ----
        if (tid < 16) {
            int p = t * 16 + tid;
            if (p < n) {
                tk[tid] = btok[e * CAP + p];
                gt[tid] = bg[e * CAP + p];
            } else {
                tk[tid] = -1;          // padded row: no output, gate = 0
                gt[tid] = 0.0f;
            }
        }
        __syncthreads();

        // ---- stage X tile into LDS: row = tid/16, 8 consecutive floats ----
        {
            int r = tid >> 4;
            int c = (tid & 15) * 8;
            int token = tk[r];
            #pragma unroll
            for (int j = 0; j < 8; ++j)
                xt[r * XPAD + c + j] =
                    (token >= 0) ? x[(size_t)token * DD + c + j] : 0.0f;
        }
        __syncthreads();

        // ---- encode: Acts[16,512] = relu(X[16,128] @ We + b_enc) * gate ----
        // 32 N-tiles of 16 columns; each wave owns 4 of them.
        for (int ntile = wave; ntile < DE / 16; ntile += 8) {
            const int n0 = ntile * 16;
            const float be = b_enc[(size_t)e * DE + n0 + ln];
            v8f acc = {be, be, be, be, be, be, be, be};
            #pragma unroll
            for (int k0 = 0; k0 < DD; k0 += 4) {
                v2f a, b;
                a.x = xt[ln * XPAD + k0 + 2 * lh];
                a.y = xt[ln * XPAD + k0 + 2 * lh + 1];
                b.x = We[(size_t)(k0 + 2 * lh) * DE + n0 + ln];
                b.y = We[(size_t)(k0 + 2 * lh + 1) * DE + n0 + ln];
                acc = __builtin_amdgcn_wmma_f32_16x16x4_f32(
                    false, a, false, b, (short)0, acc, false, false);
            }
            #pragma unroll
            for (int r = 0; r < 8; ++r) {
                int row = r + 8 * lh;
                float v = acc[r] > 0.0f ? acc[r] : 0.0f;  // relu
                acts[row * APAD + n0 + ln] = v * gt[row]; // fuse gate scale
            }
        }
        __syncthreads();

        // ---- decode: Out[16,128] = Acts[16,512] @ Wd + b_dec, scatter-add --
        // 8 N-tiles of 16 columns; one per wave. K = 512 in steps of 4.
        {
            const int n0 = wave * 16;
            const float bd = b_dec[(size_t)e * DD + n0 + ln];
            v8f acc = {bd, bd, bd, bd, bd, bd, bd, bd};
            #pragma unroll 8
            for (int k0 = 0; k0 < DE; k0 += 4) {
                v2f a, b;
                a.x = acts[ln * APAD + k0 + 2 * lh];
                a.y = acts[ln * APAD + k0 + 2 * lh + 1];
                b.x = Wd[(size_t)(k0 + 2 * lh) * DD + n0 + ln];
                b.y = Wd[(size_t)(k0 + 2 * lh + 1) * DD + n0 + ln];
                acc = __builtin_amdgcn_wmma_f32_16x16x4_f32(
                    false, a, false, b, (short)0, acc, false, false);
            }
            #pragma unroll
            for (int r = 0; r < 8; ++r) {
                int row = r + 8 * lh;
                int token = tk[row];
                if (token >= 0)
                    atomicAdd(out + (size_t)token * DD + n0 + ln, acc[r]);
            }
        }
        __syncthreads();   // protect tk/gt/acts before next tile's overwrite
    }
}

// ---------------------------------------------------------------------------
extern "C" void kernel_launch(void* const* d_in, const int* in_sizes, int n_in,
                              void* d_out, int out_size, void* d_ws, size_t ws_size,
                              hipStream_t stream) {
    const float* x     = (const float*)d_in[0];
    const float* gate  = (const float*)d_in[1];
    const float* W_enc = (const float*)d_in[2];
    const float* W_dec = (const float*)d_in[3];
    const float* b_enc = (const float*)d_in[4];
    const float* b_dec = (const float*)d_in[5];
    // d_in[6] is k (=2); routing scans for nonzeros so it is not needed.
    float* out = (float*)d_out;

    // workspace layout: counters | bucket tokens | bucket gates (~263 KB)
    int*   cnt  = (int*)d_ws;
    int*   btok = cnt + NSAE;
    float* bgv  = (float*)(btok + NSAE * CAP);

    sae_init_kernel<<<dim3((BTOT * DD + 255) / 256), 256, 0, stream>>>(out, cnt);
    sae_route_kernel<<<dim3((BTOT + 255) / 256), 256, 0, stream>>>(gate, cnt, btok, bgv);
    sae_tile_kernel<<<dim3(NSAE, 4), 256, 0, stream>>>(
        x, W_enc, W_dec, b_enc, b_dec, cnt, btok, bgv, out);
}